// TAN2dProposal_11390253269012
// MI455X (gfx1250) — compile-verified
//
#include <hip/hip_runtime.h>
#include <hip/hip_bf16.h>
#include <stdint.h>

// ---- Problem sizes (fixed by reference setup_inputs) ----
#define B_       16
#define T_       1024
#define D_       256
#define SCALE_   8
#define N_       (T_ / SCALE_)    // 128
#define NN_      (N_ * N_)        // 16384
#define WINDOW_  16
#define NEG_INF_ (-1e30f)

// Native clang vectors (usable with nontemporal builtins, unlike HIP float4).
typedef __attribute__((ext_vector_type(4))) float f32x4;
typedef __attribute__((ext_vector_type(2))) float f32x2;

// ---- gfx1250 async global->LDS path (guarded; fallback = cooperative loads) ----
#if defined(__has_builtin)
#if __has_builtin(__builtin_amdgcn_global_load_async_to_lds_b128) && \
    __has_builtin(__builtin_amdgcn_s_wait_asynccnt)
#define USE_ASYNC_LDS 1
#endif
#endif
#ifndef USE_ASYNC_LDS
#define USE_ASYNC_LDS 0
#endif

#if USE_ASYNC_LDS
// Builtin signature (probe-confirmed via hipcc diagnostic): v4i addrspace(1)*,
// v4i addrspace(3)*, imm offset, imm cpol.
typedef __attribute__((ext_vector_type(4))) int v4i_t;
typedef __attribute__((address_space(1))) v4i_t gbl_v4i;
typedef __attribute__((address_space(3))) v4i_t lds_v4i;
#endif

// -----------------------------------------------------------------------------
// Kernel 1: masked 8-way max pool along T.
//   pooled[b,n,d] = max_{k<8} (feats[b,8n+k,d] + (1-mask[b,8n+k])*NEG_INF)
//   smask[b,n]    = max_{k<8} mask[b,8n+k]
// Block handles 4 consecutive n's; thread = (n_sub = tid>>6, dvec = (tid&63)*4).
// -----------------------------------------------------------------------------
__global__ void pool_kernel(const float* __restrict__ feats,
                            const float* __restrict__ mask,
                            float* __restrict__ pooled,
                            float* __restrict__ smask) {
    const int blk = blockIdx.x;                       // b * (N/4) + n4
    const int b   = blk / (N_ / 4);
    const int n   = (blk % (N_ / 4)) * 4 + (threadIdx.x >> 6);
    const int dv  = (threadIdx.x & 63) << 2;          // d offset, float4 granular

    const float* fbase = feats + ((size_t)b * T_ + (size_t)n * SCALE_) * D_ + dv;
    const float* mbase = mask  + (size_t)b * T_ + (size_t)n * SCALE_;

    // Warm L2 for the next pooling window (lowers to global_prefetch_b8 on gfx1250).
    __builtin_prefetch(fbase + (size_t)SCALE_ * D_, 0, 1);

    f32x4 m = {-INFINITY, -INFINITY, -INFINITY, -INFINITY};
#pragma unroll
    for (int k = 0; k < SCALE_; ++k) {
        const float msk = mbase[k];
        const float add = (1.0f - msk) * NEG_INF_;
        const f32x4 f  = *(const f32x4*)(fbase + (size_t)k * D_);
        m.x = fmaxf(m.x, f.x + add);
        m.y = fmaxf(m.y, f.y + add);
        m.z = fmaxf(m.z, f.z + add);
        m.w = fmaxf(m.w, f.w + add);
    }
    // pooled stays regular-temporal: it is re-read 16x by row_kernel and fits L2.
    *(f32x4*)(pooled + (size_t)(b * N_ + n) * D_ + dv) = m;

    if ((threadIdx.x & 63) == 0) {
        float mm = mbase[0];
#pragma unroll
        for (int k = 1; k < SCALE_; ++k) mm = fmaxf(mm, mbase[k]);
        smask[b * N_ + n] = mm;
    }
}

// -----------------------------------------------------------------------------
// Kernel 2: lengths[b] = (int) sum_t mask[b,t]
// -----------------------------------------------------------------------------
__global__ void length_kernel(const float* __restrict__ mask,
                              int* __restrict__ lengths) {
    __shared__ float red[256];
    const int b = blockIdx.x;
    const int t = threadIdx.x;
    float s = 0.0f;
    for (int i = t; i < T_; i += 256) s += mask[(size_t)b * T_ + i];
    red[t] = s;
    __syncthreads();
#pragma unroll
    for (int off = 128; off > 0; off >>= 1) {
        if (t < off) red[t] += red[t + off];
        __syncthreads();
    }
    if (t == 0) lengths[b] = (int)red[0];
}

// -----------------------------------------------------------------------------
// Kernel 3: one block per (b, s) row. Async-stage pooled[b, s..s+15, :]
// (a contiguous <=16 KB chunk) into LDS via the gfx1250 async-tensor path,
// running-max along e, then stream ALL N cells of the row (zeros for invalid
// cells) with non-temporal 16 B stores, plus mask2d/bounds scalars.
// Thread = (phase p = tid>>6 over e mod 4, dvec = (tid&63)*4).
// -----------------------------------------------------------------------------
__global__ void row_kernel(const float* __restrict__ pooled,
                           const float* __restrict__ smask,
                           const int* __restrict__ lengths,
                           float* __restrict__ out_feat,
                           float* __restrict__ out_bounds,
                           float* __restrict__ out_mask) {
    const int bs = blockIdx.x;          // b*N + s
    const int b  = bs / N_;
    const int s  = bs % N_;
    const int t  = threadIdx.x;
    const int p  = t >> 6;              // e phase 0..3
    const int dv = (t & 63) << 2;       // d offset, float4 granular
    const int nwin = min(WINDOW_, N_ - s);

    __shared__ __align__(16) float win[WINDOW_ * D_];   // 16 KB staging
    __shared__ float sm_row[N_];

    const float* src = pooled + (size_t)bs * D_;        // contiguous window start
    const int chunk_bytes = nwin * D_ * 4;

#if USE_ASYNC_LDS
    // gfx1250 async path: global -> LDS, 16 B per lane per issue (ASYNCcnt).
    for (int off = t * 16; off < chunk_bytes; off += 256 * 16) {
        __builtin_amdgcn_global_load_async_to_lds_b128(
            (gbl_v4i*)((const char*)src + off),
            (lds_v4i*)((char*)win + off),
            0, 0);
    }
    __builtin_amdgcn_s_wait_asynccnt(0);
#else
    for (int i = t; i < nwin * D_; i += 256) win[i] = src[i];
#endif
    if (t < N_) sm_row[t] = smask[b * N_ + t];
    __syncthreads();

    f32x4 run = {-INFINITY, -INFINITY, -INFINITY, -INFINITY};
    float* obase = out_feat + (size_t)(b * NN_ + s * N_) * D_ + dv;

    int j = 0;  // running-max prefix progress over window slices
    for (int e = p; e < N_; e += 4) {
        f32x4 v = {0.0f, 0.0f, 0.0f, 0.0f};
        const int jj = e - s;
        if (jj >= 0 && jj < nwin) {
            while (j <= jj) {
                const f32x4 w = *(const f32x4*)&win[j * D_ + dv];
                run.x = fmaxf(run.x, w.x);
                run.y = fmaxf(run.y, w.y);
                run.z = fmaxf(run.z, w.z);
                run.w = fmaxf(run.w, w.w);
                ++j;
            }
            v = run * sm_row[e];
        }
        // Output is written once and never re-read on device: stream past caches.
        __builtin_nontemporal_store(v, (f32x4*)(obase + (size_t)e * D_));
    }

    // Scalar outputs for this row: mask2d and bounds (threads 0..N-1 -> e).
    if (t < N_) {
        const int e  = t;
        const int jj = e - s;
        const float mv = (jj >= 0 && jj < nwin) ? sm_row[e] : 0.0f;
        const size_t cell = (size_t)b * NN_ + (size_t)s * N_ + e;
        __builtin_nontemporal_store(mv, out_mask + cell);
        const int len = lengths[b];
        const int mi  = (int)mv;   // matches astype(int32) truncation for mv >= 0
        const int b0  = min(s * SCALE_, len - 1) * mi;
        const int b1  = min((e + 1) * SCALE_ - 1, len - 1) * mi;
        f32x2 bv = {(float)b0, (float)b1};
        __builtin_nontemporal_store(bv, (f32x2*)(out_bounds + 2 * cell));
    }
}

// -----------------------------------------------------------------------------
extern "C" void kernel_launch(void* const* d_in, const int* in_sizes, int n_in,
                              void* d_out, int out_size, void* d_ws, size_t ws_size,
                              hipStream_t stream) {
    (void)in_sizes; (void)n_in; (void)out_size; (void)ws_size;

    const float* feats = (const float*)d_in[0];   // [B,T,D] f32
    const float* mask  = (const float*)d_in[1];   // [B,T]   f32

    // Workspace: pooled [B,N,D] f32 | smask [B,N] f32 | lengths [B] i32
    float* pooled  = (float*)d_ws;
    float* smask   = pooled + (size_t)B_ * N_ * D_;
    int*   lengths = (int*)(smask + (size_t)B_ * N_);

    // Output regions, flat-concatenated in reference return order.
    float* out_feat   = (float*)d_out;                              // B*N*N*D
    float* out_bounds = out_feat + (size_t)B_ * NN_ * D_;           // B*N*N*2
    float* out_mask   = out_bounds + (size_t)B_ * NN_ * 2;          // B*N*N

    pool_kernel<<<dim3(B_ * (N_ / 4)), dim3(256), 0, stream>>>(feats, mask, pooled, smask);
    length_kernel<<<dim3(B_), dim3(256), 0, stream>>>(mask, lengths);
    row_kernel<<<dim3(B_ * N_), dim3(256), 0, stream>>>(pooled, smask, lengths,
                                                        out_feat, out_bounds, out_mask);
}